// Block_line4feature_68272800137804
// MI455X (gfx1250) — compile-verified
//
#include <hip/hip_runtime.h>
#include <cstdint>

// ---------------------------------------------------------------------------
// Problem geometry: x is (B=32, C=4, H=512, W=512) f32 -> 128 planes of 512x512
// ---------------------------------------------------------------------------
#define IMG_W 512
#define IMG_H 512
#define PLANE (IMG_W * IMG_H)          // 262144
#define N_PLANES 128

#define TW 128                          // tile width (outputs)
#define TH 16                           // tile height (outputs)
#define LDS_W 136                       // TW + 8 (16B-aligned halo: 4 left, 4 right)
#define LDS_H 18                        // TH + 2
#define CHUNKS_PER_ROW 34               // LDS_W / 4 floats per b128 chunk
#define NCHUNK (CHUNKS_PER_ROW * LDS_H) // 612
#define TILES_X (IMG_W / TW)            // 4
#define TILES_Y (IMG_H / TH)            // 32
#define TILES_PER_PLANE (TILES_X * TILES_Y) // 128
#define NBLOCKS (N_PLANES * TILES_PER_PLANE) // 16384

typedef float v4f __attribute__((ext_vector_type(4)));

// ---------------------------------------------------------------------------
// Async tile loader: CDNA5 GLOBAL_LOAD_ASYNC_TO_LDS_B128 (ASYNCcnt path).
// Each lane DMAs one 16B chunk straight into LDS; out-of-image halo chunks are
// zero-filled via normal DS stores. s_wait_asynccnt 0 + barrier before use.
// ---------------------------------------------------------------------------
__device__ __forceinline__ void load_tile_async(const float* __restrict__ src,
                                                int x0, int y0,
                                                float* tile) {
  const unsigned lds_base = (unsigned)(uintptr_t)tile; // low 32 bits = LDS offset
  for (int q = threadIdx.x; q < NCHUNK; q += 256) {
    const int row = q / CHUNKS_PER_ROW;
    const int cc  = q - row * CHUNKS_PER_ROW;
    const int gy  = y0 - 1 + row;
    const int gx  = x0 - 4 + (cc << 2);
    const int lidx = row * LDS_W + (cc << 2);
    if (gy >= 0 && gy < IMG_H && gx >= 0 && gx <= IMG_W - 4) {
      const float* g = src + (size_t)gy * IMG_W + gx;        // 16B aligned
      const unsigned lds_addr = lds_base + (unsigned)(lidx * 4);
      asm volatile("global_load_async_to_lds_b128 %0, %1, off"
                   :: "v"(lds_addr), "v"(g)
                   : "memory");
    } else {
      *reinterpret_cast<v4f*>(tile + lidx) = (v4f)0.0f;
    }
  }
  asm volatile("s_wait_asynccnt 0" ::: "memory");
  __syncthreads();
}

// Fused 3x3 stencil (all four line kernels + affine collapsed; +0.5 dropped
// since InstanceNorm cancels any constant offset).
__device__ __forceinline__ float conv_at(const float* tile, int oy, int ox) {
  const float* p = tile + (oy + 1) * LDS_W + (ox + 4);
  const float c  = p[0];
  const float l  = p[-1],          r  = p[1];
  const float u  = p[-LDS_W],      d  = p[LDS_W];
  const float ul = p[-LDS_W - 1],  ur = p[-LDS_W + 1];
  const float dl = p[LDS_W - 1],   dr = p[LDS_W + 1];
  return 0.5f * c
       - (1.0f / 60.0f) * (l + r)
       - (1.0f / 15.0f) * (u + d)
       - (1.0f / 30.0f) * (ur + dl)
       - (2.0f / 15.0f) * (ul + dr);
}

// Compute 4 consecutive outputs of one tile row as a vector.
__device__ __forceinline__ v4f conv4_at(const float* tile, int oy, int ox) {
  v4f z;
  z.x = conv_at(tile, oy, ox + 0);
  z.y = conv_at(tile, oy, ox + 1);
  z.z = conv_at(tile, oy, ox + 2);
  z.w = conv_at(tile, oy, ox + 3);
  return z;
}

// ---------------------------------------------------------------------------
// Pass 1: per-plane sum / sum-of-squares of the stencil output.
// acc[0..127] = sum, acc[128..255] = sumsq (pre-zeroed via hipMemsetAsync).
// ---------------------------------------------------------------------------
__global__ __launch_bounds__(256) void stats_kernel(const float* __restrict__ x,
                                                    float* __restrict__ acc) {
  __shared__ __align__(16) float tile[LDS_H * LDS_W];
  __shared__ float red[2];
  const int plane = blockIdx.x / TILES_PER_PLANE;
  const int t     = blockIdx.x - plane * TILES_PER_PLANE;
  const int x0    = (t % TILES_X) * TW;
  const int y0    = (t / TILES_X) * TH;
  const float* src = x + (size_t)plane * PLANE;

  if (threadIdx.x == 0) { red[0] = 0.f; red[1] = 0.f; }
  load_tile_async(src, x0, y0, tile);   // includes __syncthreads()

  float s1 = 0.f, s2 = 0.f;
#pragma unroll
  for (int k = 0; k < 2; ++k) {
    const int chunk = threadIdx.x + (k << 8);   // 0..511 (tile = 512 x float4)
    const int oy = chunk >> 5;                  // 0..15
    const int ox = (chunk & 31) << 2;           // 0,4,...,124
    const v4f z = conv4_at(tile, oy, ox);
    s1 += z.x + z.y + z.z + z.w;
    s2 += z.x * z.x + z.y * z.y + z.z * z.z + z.w * z.w;
  }
  // wave32 shuffle reduction
#pragma unroll
  for (int off = 16; off > 0; off >>= 1) {
    s1 += __shfl_down(s1, off, 32);
    s2 += __shfl_down(s2, off, 32);
  }
  if ((threadIdx.x & 31) == 0) {
    atomicAdd(&red[0], s1);
    atomicAdd(&red[1], s2);
  }
  __syncthreads();
  if (threadIdx.x == 0) {
    atomicAdd(&acc[plane], red[0]);
    atomicAdd(&acc[N_PLANES + plane], red[1]);
  }
}

// ---------------------------------------------------------------------------
// Pass 2: per-plane mean / rsqrt(var + eps).  ws[256..383]=mean, ws[384..511]=inv
// ---------------------------------------------------------------------------
__global__ void finalize_kernel(float* __restrict__ ws) {
  const int p = threadIdx.x;
  if (p < N_PLANES) {
    const float inv_n = 1.0f / (float)PLANE;
    const float mean  = ws[p] * inv_n;
    float var = ws[N_PLANES + p] * inv_n - mean * mean;
    var = fmaxf(var, 0.0f);
    ws[2 * N_PLANES + p] = mean;
    ws[3 * N_PLANES + p] = rsqrtf(var + 1e-5f);
  }
}

// ---------------------------------------------------------------------------
// Pass 3: recompute stencil (input is L2-resident: 128 MiB < 192 MB L2),
// normalize, coalesced non-temporal B128 stores (keep input hot in L2).
// ---------------------------------------------------------------------------
__global__ __launch_bounds__(256) void norm_kernel(const float* __restrict__ x,
                                                   const float* __restrict__ stats,
                                                   float* __restrict__ out) {
  __shared__ __align__(16) float tile[LDS_H * LDS_W];
  const int plane = blockIdx.x / TILES_PER_PLANE;
  const int t     = blockIdx.x - plane * TILES_PER_PLANE;
  const int x0    = (t % TILES_X) * TW;
  const int y0    = (t / TILES_X) * TH;
  const float* src = x + (size_t)plane * PLANE;
  const float mean = stats[plane];
  const float inv  = stats[N_PLANES + plane];

  load_tile_async(src, x0, y0, tile);

  float* dst = out + (size_t)plane * PLANE;
#pragma unroll
  for (int k = 0; k < 2; ++k) {
    const int chunk = threadIdx.x + (k << 8);   // 0..511
    const int oy = chunk >> 5;
    const int ox = (chunk & 31) << 2;
    v4f z = conv4_at(tile, oy, ox);
    z.x = (z.x - mean) * inv;
    z.y = (z.y - mean) * inv;
    z.z = (z.z - mean) * inv;
    z.w = (z.w - mean) * inv;
    v4f* addr = reinterpret_cast<v4f*>(dst + (size_t)(y0 + oy) * IMG_W + (x0 + ox));
    __builtin_nontemporal_store(z, addr);       // th:NT — don't evict input from L2
  }
}

// ---------------------------------------------------------------------------
extern "C" void kernel_launch(void* const* d_in, const int* in_sizes, int n_in,
                              void* d_out, int out_size, void* d_ws, size_t ws_size,
                              hipStream_t stream) {
  (void)in_sizes; (void)n_in; (void)out_size; (void)ws_size;
  const float* x  = (const float*)d_in[0];
  float* out      = (float*)d_out;
  float* ws       = (float*)d_ws;           // uses 512 floats (2 KB)

  // zero the per-plane accumulators (graph-capture-safe memset node)
  hipMemsetAsync(ws, 0, 2 * N_PLANES * sizeof(float), stream);

  stats_kernel<<<NBLOCKS, 256, 0, stream>>>(x, ws);
  finalize_kernel<<<1, N_PLANES, 0, stream>>>(ws);
  norm_kernel<<<NBLOCKS, 256, 0, stream>>>(x, ws + 2 * N_PLANES, out);
}